// ISED_73005854097950
// MI455X (gfx1250) — compile-verified
//
#include <hip/hip_runtime.h>
#include <hip/hip_bf16.h>

// Problem shape (fixed by reference)
#define B_DIM 4096
#define K_DIM 1024
#define C_DIM 10
#define M_DIM 19

// Count kernel config
#define TPB 256
#define NCHUNK 8
#define KCHUNK (K_DIM / NCHUNK)   // 128 samples per chunk -> max cell count 128 (fits u16)
#define CWORDS 51                 // 100 cells packed 2-per-u32 + 1 pad; odd stride => conflict-free LDS banks

// Normalize kernel config
#define NWAVES_NORM 4

typedef float v2f __attribute__((ext_vector_type(2)));
typedef float v8f __attribute__((ext_vector_type(8)));

#if defined(__has_builtin)
#if __has_builtin(__builtin_amdgcn_wmma_f32_16x16x4_f32)
#define HAVE_WMMA_F32X4 1
#endif
#endif

__global__ void ISED_zero_kernel(float* __restrict__ out, int n) {
    int i = blockIdx.x * blockDim.x + threadIdx.x;
    if (i < n) out[i] = 0.0f;
}

// One thread per batch row b (lanes span consecutive b => every wave iteration
// consumes exactly one 128B line of each idx stream). Inner loop: 2 coalesced
// b32 loads + v_mad + private ds_add_u32. All float math deferred to the flush.
__global__ __launch_bounds__(TPB) void ISED_count_scatter_kernel(
    const float* __restrict__ x1, const float* __restrict__ x2,
    const int* __restrict__ idx1, const int* __restrict__ idx2,
    float* __restrict__ out) {
    __shared__ unsigned int cnt[TPB * CWORDS];   // 52,224 B

    const int t = threadIdx.x;
    const int b = blockIdx.x * TPB + t;
    unsigned int* my = &cnt[t * CWORDS];         // strictly private region: no barriers needed

#pragma unroll
    for (int i = 0; i < CWORDS; ++i) my[i] = 0u;

    const size_t base = (size_t)(blockIdx.y * KCHUNK) * B_DIM + b;
    const int* p1 = idx1 + base;
    const int* p2 = idx2 + base;

#pragma unroll 8
    for (int k = 0; k < KCHUNK; ++k) {
        int i1 = p1[(size_t)k * B_DIM];                 // coalesced across lanes
        int i2 = p2[(size_t)k * B_DIM];
        int cell = i1 * C_DIM + i2;                     // 0..99
        unsigned int inc = 1u << ((cell & 1) << 4);     // packed u16 halves, no carry possible
        atomicAdd(&my[cell >> 1], inc);                 // ds_add_u32 (no return), private addr
    }

    // Flush: 10x10 counts -> 19 float bins (fully unrolled, static register indexing)
    float xa[C_DIM], xb[C_DIM];
#pragma unroll
    for (int c = 0; c < C_DIM; ++c) {
        xa[c] = x1[b * C_DIM + c];
        xb[c] = x2[b * C_DIM + c];
    }
    float bins[M_DIM];
#pragma unroll
    for (int m = 0; m < M_DIM; ++m) bins[m] = 0.0f;
#pragma unroll
    for (int c1 = 0; c1 < C_DIM; ++c1) {
#pragma unroll
        for (int c2 = 0; c2 < C_DIM; ++c2) {
            int cell = c1 * C_DIM + c2;
            unsigned int w = my[cell >> 1];                          // CSE'd pairwise
            float cf = (float)((w >> ((cell & 1) << 4)) & 0xFFFFu);  // exact integer count
            bins[c1 + c2] = fmaf(xa[c1] * xb[c2], cf, bins[c1 + c2]);
        }
    }

    float* row = out + (size_t)b * M_DIM;
#pragma unroll
    for (int m = 0; m < M_DIM; ++m) {
        unsafeAtomicAdd(&row[m], bins[m]);   // global_atomic_add_f32
    }
}

// L2-normalize rows. Sum-of-squares per row computed as the diagonal of
// D = V * V^T via V_WMMA_F32_16X16X4_F32 (exact f32 math), one wave per 16 rows.
// A-matrix 16x4 layout (ISA 7.12.2): lanes0-15 hold (M=lane, K=0/1 in v0/v1),
// lanes16-31 hold (M=lane-16, K=2/3). B-matrix 4x16 has the same register image
// for V*V^T, so the same registers feed both operands.
__global__ __launch_bounds__(32 * NWAVES_NORM) void ISED_normalize_kernel(float* __restrict__ out) {
#ifdef HAVE_WMMA_F32X4
    const int lane = threadIdx.x & 31;
    const int wave = threadIdx.x >> 5;
    const int b0 = (blockIdx.x * NWAVES_NORM + wave) * 16;
    const int r  = lane & 15;     // tile row this lane services
    const int hi = lane >> 4;     // 0 -> K=0,1 ; 1 -> K=2,3
    const float* rowp = out + (size_t)(b0 + r) * M_DIM;

    v8f acc = {0.f, 0.f, 0.f, 0.f, 0.f, 0.f, 0.f, 0.f};
#pragma unroll
    for (int chunk = 0; chunk < 5; ++chunk) {         // 5*4 = 20 >= M_DIM (pad with 0)
        int k0 = chunk * 4 + hi * 2;
        v2f a;
        a[0] = (k0 + 0 < M_DIM) ? rowp[k0 + 0] : 0.0f;
        a[1] = (k0 + 1 < M_DIM) ? rowp[k0 + 1] : 0.0f;
        acc = __builtin_amdgcn_wmma_f32_16x16x4_f32(
            /*neg_a=*/false, a, /*neg_b=*/false, a,
            /*c_mod=*/(short)0, acc, /*reuse_a=*/false, /*reuse_b=*/false);
    }
    // D layout: acc[v] = D[v][lane] for lanes0-15, D[v+8][lane-16] for lanes16-31.
    // Diagonal D[i][i]: i<8 at lane i / acc[i]; i>=8 at lane i+16 / acc[i-8].
    int j = lane & 7;
    float md = acc[0];
    md = (j == 1) ? acc[1] : md;
    md = (j == 2) ? acc[2] : md;
    md = (j == 3) ? acc[3] : md;
    md = (j == 4) ? acc[4] : md;
    md = (j == 5) ? acc[5] : md;
    md = (j == 6) ? acc[6] : md;
    md = (j == 7) ? acc[7] : md;
    int src = (r < 8) ? r : (r + 8);
    float norm2 = __shfl(md, src, 32);                // ds_bpermute broadcast
    float inv = 1.0f / fmaxf(sqrtf(norm2), 1e-12f);

    float* wr = out + (size_t)(b0 + r) * M_DIM;
    if (hi == 0) {
#pragma unroll
        for (int m = 0; m < 10; ++m) wr[m] = wr[m] * inv;
    } else {
#pragma unroll
        for (int m = 10; m < M_DIM; ++m) wr[m] = wr[m] * inv;
    }
#else
    // Scalar fallback (same launch geometry: 8192 threads cover 4096 rows)
    int b = blockIdx.x * blockDim.x + threadIdx.x;
    if (b < B_DIM) {
        float* wr = out + (size_t)b * M_DIM;
        float s = 0.0f;
#pragma unroll
        for (int m = 0; m < M_DIM; ++m) s = fmaf(wr[m], wr[m], s);
        float inv = 1.0f / fmaxf(sqrtf(s), 1e-12f);
#pragma unroll
        for (int m = 0; m < M_DIM; ++m) wr[m] = wr[m] * inv;
    }
#endif
}

extern "C" void kernel_launch(void* const* d_in, const int* in_sizes, int n_in,
                              void* d_out, int out_size, void* d_ws, size_t ws_size,
                              hipStream_t stream) {
    (void)in_sizes; (void)n_in; (void)d_ws; (void)ws_size;
    const float* x1  = (const float*)d_in[0];
    const float* x2  = (const float*)d_in[1];
    const int* idx1  = (const int*)d_in[2];
    const int* idx2  = (const int*)d_in[3];
    float* out = (float*)d_out;

    // 1) zero the accumulator (harness poisons d_out)
    ISED_zero_kernel<<<(out_size + 255) / 256, 256, 0, stream>>>(out, out_size);

    // 2) count + scatter-add (grid: 16 b-tiles x 8 k-chunks = 128 blocks, 1024 waves)
    dim3 grid(B_DIM / TPB, NCHUNK);
    ISED_count_scatter_kernel<<<grid, TPB, 0, stream>>>(x1, x2, idx1, idx2, out);

    // 3) L2 normalize rows (WMMA V*V^T for the norm reduction)
    ISED_normalize_kernel<<<B_DIM / (16 * NWAVES_NORM), 32 * NWAVES_NORM, 0, stream>>>(out);
}